// GraphSAGE_5411658793415
// MI455X (gfx1250) — compile-verified
//
#include <hip/hip_runtime.h>
#include <cstdint>

typedef __attribute__((ext_vector_type(2))) float v2f;
typedef __attribute__((ext_vector_type(8))) float v8f;

#define NODES   50000
#define THREADS 256

// ---------------------------------------------------------------- utilities
__global__ void zero_f32(float* __restrict__ p, size_t n) {
    size_t i = (size_t)blockIdx.x * blockDim.x + threadIdx.x;
    size_t stride = (size_t)gridDim.x * blockDim.x;
    for (; i < n; i += stride) p[i] = 0.0f;
}

__global__ void deg_count(const long long* __restrict__ dst, float* __restrict__ deg, int E) {
    int e = blockIdx.x * blockDim.x + threadIdx.x;
    if (e < E) unsafeAtomicAdd(&deg[dst[e]], 1.0f);
}

__global__ void rdeg_kernel(float* __restrict__ deg, int n) {
    int i = blockIdx.x * blockDim.x + threadIdx.x;
    if (i < n) deg[i] = 1.0f / fmaxf(deg[i], 1.0f);
}

// One edge per block, one channel per thread. Gather h[src] row, atomically
// accumulate into agg[dst] row (global_atomic_add_f32, wait-free HW atomic).
// Prefetch the next edge's source row to hide gather latency.
__global__ void scatter_add(const long long* __restrict__ src,
                            const long long* __restrict__ dst,
                            const float* __restrict__ h,
                            float* __restrict__ agg,
                            int E, int C) {
    int e = blockIdx.x;
    int c = threadIdx.x;
    long long s = src[e];
    long long d = dst[e];
    float v = h[(size_t)s * C + c];
    if (e + 1 < E) {
        long long s2 = src[e + 1];
        __builtin_prefetch(&h[(size_t)s2 * C + c], 0, 1);  // global_prefetch_b8
    }
    unsafeAtomicAdd(&agg[(size_t)d * C + c], v);
}

// ---------------------------------------------------------------- fused SAGE GEMM
// out[M,NOUT] = (agg ∘ rdeg) @ Wl + h @ Wr + bias   (optional ReLU)
// One 16x16 tile per wave via V_WMMA_F32_16X16X4_F32, K-step 4.
// Block = 256 threads = 8 waves; waves tile the N dimension, blockIdx.x tiles M.
template <int K, int NOUT, bool RELU>
__global__ __launch_bounds__(THREADS)
void sage_gemm(const float* __restrict__ agg,
               const float* __restrict__ rdeg,
               const float* __restrict__ h,
               const float* __restrict__ Wl,
               const float* __restrict__ Wr,
               const float* __restrict__ bias,
               float* __restrict__ out) {
    const int lane = threadIdx.x & 31;
    const int wv   = threadIdx.x >> 5;
    const int half = lane >> 4;    // 0: lanes 0-15, 1: lanes 16-31
    const int l16  = lane & 15;

    const int m0 = blockIdx.x * 16;
    const int n0 = (blockIdx.y * (THREADS >> 5) + wv) * 16;  // always < NOUT by launch config

    const int row = m0 + l16;       // A-fragment row for this lane
    const int col = n0 + l16;       // B/C/D column for this lane
    const float rs = rdeg[row];

    // C init: broadcast bias[col] into all 8 accumulator VGPRs
    v8f acc;
    {
        const float bv = bias[col];
#pragma unroll
        for (int g = 0; g < 8; ++g) acc[g] = bv;
    }

    const float* __restrict__ arow = agg + (size_t)row * K;
    const float* __restrict__ hrow = h   + (size_t)row * K;

    // --- pass 1: (agg * rdeg) @ Wl -------------------------------------
#pragma unroll 4
    for (int k0 = 0; k0 < K; k0 += 4) {
        const int ka = k0 + 2 * half;              // lane's K pair
        v2f a; a.x = arow[ka] * rs; a.y = arow[ka + 1] * rs;
        v2f b; b.x = Wl[(size_t)ka * NOUT + col];
               b.y = Wl[(size_t)(ka + 1) * NOUT + col];
        acc = __builtin_amdgcn_wmma_f32_16x16x4_f32(
            /*neg_a=*/false, a, /*neg_b=*/false, b,
            /*c_mod=*/(short)0, acc, /*reuse_a=*/false, /*reuse_b=*/false);
    }
    // --- pass 2: h @ Wr -------------------------------------------------
#pragma unroll 4
    for (int k0 = 0; k0 < K; k0 += 4) {
        const int ka = k0 + 2 * half;
        v2f a; a.x = hrow[ka]; a.y = hrow[ka + 1];
        v2f b; b.x = Wr[(size_t)ka * NOUT + col];
               b.y = Wr[(size_t)(ka + 1) * NOUT + col];
        acc = __builtin_amdgcn_wmma_f32_16x16x4_f32(
            false, a, false, b, (short)0, acc, false, false);
    }

    // --- store: VGPR g -> row m0 + g + 8*half, column col ---------------
    const int mstore = m0 + 8 * half;
#pragma unroll
    for (int g = 0; g < 8; ++g) {
        float v = acc[g];
        if (RELU) v = fmaxf(v, 0.0f);
        out[(size_t)(mstore + g) * NOUT + col] = v;
    }
}

// ---------------------------------------------------------------- launcher
extern "C" void kernel_launch(void* const* d_in, const int* in_sizes, int n_in,
                              void* d_out, int out_size, void* d_ws, size_t ws_size,
                              hipStream_t stream) {
    const float*     x    = (const float*)d_in[0];
    const long long* ei   = (const long long*)d_in[1];   // int64 [2, E]
    const float*     Wl1  = (const float*)d_in[2];
    const float*     Wr1  = (const float*)d_in[3];
    const float*     b1   = (const float*)d_in[4];
    const float*     Wl2  = (const float*)d_in[5];
    const float*     Wr2  = (const float*)d_in[6];
    const float*     b2   = (const float*)d_in[7];
    float*           out  = (float*)d_out;

    const int N = NODES;              // 50000 = 16 * 3125
    const int E = in_sizes[1] / 2;    // 640000
    const long long* src = ei;
    const long long* dst = ei + E;

    // workspace layout (floats): deg[N] | agg1[N*128] | h1[N*256] | agg2[N*256]
    float* ws   = (float*)d_ws;
    float* deg  = ws;
    float* agg1 = deg  + N;
    float* h1   = agg1 + (size_t)N * 128;
    float* agg2 = h1   + (size_t)N * 256;

    const int ZG = 4096;

    // -------- layer 1 --------
    zero_f32<<<ZG, THREADS, 0, stream>>>(deg, (size_t)N * (1 + 128)); // deg + agg1 contiguous
    deg_count<<<(E + THREADS - 1) / THREADS, THREADS, 0, stream>>>(dst, deg, E);
    scatter_add<<<E, 128, 0, stream>>>(src, dst, x, agg1, E, 128);
    rdeg_kernel<<<(N + THREADS - 1) / THREADS, THREADS, 0, stream>>>(deg, N);

    // h1 = relu((agg1∘rdeg) @ Wl1 + x @ Wr1 + b1) : M=50000, K=128, NOUT=256
    sage_gemm<128, 256, true><<<dim3(N / 16, 2), THREADS, 0, stream>>>(
        agg1, deg, x, Wl1, Wr1, b1, h1);

    // -------- layer 2 --------
    zero_f32<<<ZG, THREADS, 0, stream>>>(agg2, (size_t)N * 256);
    scatter_add<<<E, 256, 0, stream>>>(src, dst, h1, agg2, E, 256);

    // out = (agg2∘rdeg) @ Wl2 + h1 @ Wr2 + b2 : K=256, NOUT=128
    sage_gemm<256, 128, false><<<dim3(N / 16, 1), THREADS, 0, stream>>>(
        agg2, deg, h1, Wl2, Wr2, b2, out);
}